// RCFLoss_69552700391741
// MI455X (gfx1250) — compile-verified
//
#include <hip/hip_runtime.h>
#include <hip/hip_bf16.h>
#include <math.h>

// ---------------------------------------------------------------------------
// RCF edge loss, fused single-pass tile kernel for MI455X (gfx1250, wave32).
//
// Memory-bound problem (~34MB mandatory reads @ 23.3TB/s => ~1.4us floor).
// Strategy: one fused kernel, 32x32 output tile with 8-pixel halo (48x48),
// everything in LDS; separable 9x9 box sums done as banded 16x16 GEMM blocks
// on V_WMMA_F32_16X16X4_F32 (matrix pipes are otherwise idle). Per-block
// partial sums -> workspace -> deterministic finalize kernel.
// ---------------------------------------------------------------------------

typedef __attribute__((ext_vector_type(2))) float v2f;
typedef __attribute__((ext_vector_type(8))) float v8f;

#define IMG   512
#define TILE  32
#define PADH  48          // TILE + 2*8 halo
#define LDP   49          // LDS row stride (odd -> conflict-free column access)
#define B9STR 33          // stride for 32x32 inner-result buffers

// ---------------------------------------------------------------------------
// Row-direction radius-4 box sum of one 16x16 block via WMMA:
//   D[m][n] = sum_{dx=-4..4} S[row0+m][col0+n+dx]
// out(16x16) = A(16x24) x W(24x16), W[j][n] = (n <= j <= n+8), done as six
// K=4 chunks on V_WMMA_F32_16X16X4_F32.
// A layout (ISA 7.12.2, 32-bit A 16x4): lane L holds M=L&15; VGPR pair holds
// K = 2*(L>>4) and K+1.  B 4x16: VGPR i holds rows i / i+2 striped over lanes.
// ---------------------------------------------------------------------------
__device__ __forceinline__ void wmma_box9_row(const float* S, float* D,
                                              int row0, int col0, int lane) {
    const int m    = lane & 15;   // A row / B col / D col
    const int half = lane >> 4;   // 0 or 1
    v8f acc = {};
    const float* src = &S[(row0 + m) * LDP + (col0 - 4)];
#pragma unroll
    for (int c = 0; c < 6; ++c) {
        const int jk = 4 * c + 2 * half;     // source column offset (0..23)
        v2f a, b;
        a.x = src[jk];
        a.y = src[jk + 1];
        b.x = (jk     >= m && jk     <= m + 8) ? 1.0f : 0.0f;
        b.y = (jk + 1 >= m && jk + 1 <= m + 8) ? 1.0f : 0.0f;
        acc = __builtin_amdgcn_wmma_f32_16x16x4_f32(
            false, a, false, b, (short)0, acc, false, false);
    }
#pragma unroll
    for (int v = 0; v < 8; ++v)
        D[(row0 + v + 8 * half) * LDP + col0 + m] = acc[v];
}

// ---------------------------------------------------------------------------
// Column-direction radius-4 box sum of one 16x16 block via WMMA:
//   D[m][n] = sum_{dy=-4..4} S[row0+m+dy][col0+n]
// out = W'(16x24) x B(24x16), W'[m][j] = (m <= j <= m+8).
// Destination is parameterized (stride/offset) so results can land either in
// a full 48-wide LDS field or a packed 32x32 inner buffer.
// ---------------------------------------------------------------------------
__device__ __forceinline__ void wmma_box9_col(const float* S, float* D,
                                              int row0, int col0,
                                              int dstr, int dro, int dco,
                                              int lane) {
    const int m    = lane & 15;
    const int half = lane >> 4;
    v8f acc = {};
#pragma unroll
    for (int c = 0; c < 6; ++c) {
        const int jk = 4 * c + 2 * half;     // source row offset (0..23)
        v2f a, b;
        a.x = (jk     >= m && jk     <= m + 8) ? 1.0f : 0.0f;
        a.y = (jk + 1 >= m && jk + 1 <= m + 8) ? 1.0f : 0.0f;
        b.x = S[(row0 - 4 + jk)     * LDP + col0 + m];
        b.y = S[(row0 - 4 + jk + 1) * LDP + col0 + m];
        acc = __builtin_amdgcn_wmma_f32_16x16x4_f32(
            false, a, false, b, (short)0, acc, false, false);
    }
#pragma unroll
    for (int v = 0; v < 8; ++v)
        D[(row0 + v + 8 * half - dro) * dstr + (col0 + m - dco)] = acc[v];
}

// ---------------------------------------------------------------------------
// Fused tile kernel. Grid: (16,16,16) = (x tiles, y tiles, images), 256 thr.
// Writes 6 partial sums per block: {S0, S1, num_pos, num_neg, tex, bdr}.
// ---------------------------------------------------------------------------
__global__ __launch_bounds__(256) void rcf_tile_kernel(
    const float* __restrict__ inp, const float* __restrict__ tgt,
    float* __restrict__ partial) {

    __shared__ float sP [PADH * LDP];   // p
    __shared__ float sL [PADH * LDP];   // label_w
    __shared__ float sR [PADH * LDP];   // row-sum scratch
    __shared__ float sTM[PADH * LDP];   // boxsum9(label_w)
    __shared__ float sW [PADH * LDP];   // p*label_w, later q
    __shared__ float sB9[TILE * B9STR]; // boxsum9(p*label_w) inner
    __shared__ float sPT[TILE * B9STR]; // boxsum9(q) inner
    __shared__ float sRed[6 * 256];

    const int tid  = threadIdx.x;
    const int lane = tid & 31;
    const int wave = tid >> 5;
    const int gx0  = blockIdx.x * TILE - 8;
    const int gy0  = blockIdx.y * TILE - 8;
    const float* p  = inp + (size_t)blockIdx.z * IMG * IMG;
    const float* lb = tgt + (size_t)blockIdx.z * IMG * IMG;

    float s0 = 0.f, s1 = 0.f, np = 0.f, nn = 0.f, tex = 0.f, bdr = 0.f;

    // ---- load halo'd tile, derive label_w, accumulate BCE terms (inner) ----
    for (int i = tid; i < PADH * PADH; i += 256) {
        const int r = i / PADH, c = i % PADH;
        const int gy = gy0 + r, gx = gx0 + c;
        float pv = 0.f, lv = 0.f;
        if (gy >= 0 && gy < IMG && gx >= 0 && gx < IMG) {
            pv = p[gy * IMG + gx];
            lv = lb[gy * IMG + gx];
        }
        sP[r * LDP + c] = pv;
        sL[r * LDP + c] = (lv != 0.0f) ? 1.0f : 0.0f;
        if (r >= 8 && r < 8 + TILE && c >= 8 && c < 8 + TILE) {
            if (lv == 1.0f) { s1 -= fmaxf(logf(pv),        -100.0f); np += 1.f; }
            else if (lv == 0.0f) { s0 -= fmaxf(logf(1.0f - pv), -100.0f); nn += 1.f; }
        }
    }
    __syncthreads();

    // ---- stage 1: sTM = boxsum9(label_w) on [4..43]^2 ----
    {
        const int rA[3] = {0, 16, 32}, cA[3] = {4, 20, 28};
        for (int b = wave; b < 9; b += 8)
            wmma_box9_row(sL, sR, rA[b / 3], cA[b % 3], lane);
    }
    __syncthreads();
    {
        const int rB[3] = {4, 20, 28};
        for (int b = wave; b < 9; b += 8)
            wmma_box9_col(sR, sTM, rB[b / 3], rB[b % 3], LDP, 0, 0, lane);
    }
    __syncthreads();

    // ---- stage 2: sB9 = boxsum9(p * label_w) on inner 32x32 ----
    for (int i = tid; i < PADH * PADH; i += 256) {
        const int r = i / PADH, c = i % PADH;
        sW[r * LDP + c] = sP[r * LDP + c] * sL[r * LDP + c];
    }
    __syncthreads();
    {
        const int rA[3] = {4, 20, 28}, cA[2] = {8, 24};
        for (int b = wave; b < 6; b += 8)
            wmma_box9_row(sW, sR, rA[b / 2], cA[b % 2], lane);
    }
    __syncthreads();
    {
        const int rc[2] = {8, 24};
        for (int b = wave; b < 4; b += 8)
            wmma_box9_col(sR, sB9, rc[b / 2], rc[b % 2], B9STR, 8, 8, lane);
    }
    __syncthreads();

    // ---- stage 3: q = p*(1-label_w)*(TM>0); sPT = boxsum9(q) inner ----
    for (int i = tid; i < PADH * PADH; i += 256) {
        const int r = i / PADH, c = i % PADH;
        if (r >= 4 && r < 44 && c >= 4 && c < 44) {
            const float lw = sL[r * LDP + c];
            const float bm = (sTM[r * LDP + c] > 0.0f) ? 1.0f : 0.0f;
            sW[r * LDP + c] = sP[r * LDP + c] * (1.0f - lw) * bm;
        }
    }
    __syncthreads();
    {
        const int rA[3] = {4, 20, 28}, cA[2] = {8, 24};
        for (int b = wave; b < 6; b += 8)
            wmma_box9_row(sW, sR, rA[b / 2], cA[b % 2], lane);
    }
    __syncthreads();
    {
        const int rc[2] = {8, 24};
        for (int b = wave; b < 4; b += 8)
            wmma_box9_col(sR, sPT, rc[b / 2], rc[b % 2], B9STR, 8, 8, lane);
    }
    __syncthreads();

    // ---- stage 4: per-pixel texture + boundary losses on inner 32x32 ----
    for (int i = tid; i < TILE * TILE; i += 256) {
        const int r = 8 + (i >> 5), c = 8 + (i & 31);
        const float Lv  = sL[r * LDP + c];
        const float TMv = sTM[r * LDP + c];

        float ps3 = 0.f;                       // 3x3 box sum of p
#pragma unroll
        for (int dy = -1; dy <= 1; ++dy)
#pragma unroll
            for (int dx = -1; dx <= 1; ++dx)
                ps3 += sP[(r + dy) * LDP + (c + dx)];

        float om = 1.0f - ps3 * (1.0f / 9.0f);
        om = fminf(fmaxf(om, 1e-10f), 1.0f - 1e-10f);
        if (TMv <= 0.0f) tex -= logf(om);      // tmask = 1 where boxsum9(L)==0

        if (Lv != 0.0f) {
            const float pb = Lv * sB9[(r - 8) * B9STR + (c - 8)];
            const float pt = sPT[(r - 8) * B9STR + (c - 8)];
            float sm = pb / (pt + pb + 1e-10f);
            sm = fminf(fmaxf(sm, 1e-10f), 1.0f - 1e-10f);
            bdr -= Lv * logf(sm);
        }
    }

    // ---- block reduction of the 6 scalars ----
    sRed[0 * 256 + tid] = s0;
    sRed[1 * 256 + tid] = s1;
    sRed[2 * 256 + tid] = np;
    sRed[3 * 256 + tid] = nn;
    sRed[4 * 256 + tid] = tex;
    sRed[5 * 256 + tid] = bdr;
    __syncthreads();
    for (int s = 128; s > 0; s >>= 1) {
        if (tid < s)
#pragma unroll
            for (int k = 0; k < 6; ++k)
                sRed[k * 256 + tid] += sRed[k * 256 + tid + s];
        __syncthreads();
    }
    if (tid < 6) {
        const int bid = (blockIdx.z * gridDim.y + blockIdx.y) * gridDim.x + blockIdx.x;
        partial[bid * 8 + tid] = sRed[tid * 256];
    }
}

// ---------------------------------------------------------------------------
// Deterministic finalize: reduce per-block partials, combine loss terms.
// ---------------------------------------------------------------------------
__global__ __launch_bounds__(256) void rcf_finalize(
    const float* __restrict__ partial, int nblocks, float* __restrict__ out) {
    __shared__ float red[6 * 256];
    float acc[6] = {0.f, 0.f, 0.f, 0.f, 0.f, 0.f};
    for (int b = threadIdx.x; b < nblocks; b += 256)
#pragma unroll
        for (int k = 0; k < 6; ++k) acc[k] += partial[b * 8 + k];
#pragma unroll
    for (int k = 0; k < 6; ++k) red[k * 256 + threadIdx.x] = acc[k];
    __syncthreads();
    for (int s = 128; s > 0; s >>= 1) {
        if (threadIdx.x < s)
#pragma unroll
            for (int k = 0; k < 6; ++k)
                red[k * 256 + threadIdx.x] += red[k * 256 + threadIdx.x + s];
        __syncthreads();
    }
    if (threadIdx.x == 0) {
        const float S0  = red[0 * 256];
        const float S1  = red[1 * 256];
        const float np  = red[2 * 256];
        const float nn  = red[3 * 256];
        const float tex = red[4 * 256];
        const float bdr = red[5 * 256];
        const float tot = np + nn;
        const float cost = (nn / tot) * S1 + 1.1f * (np / tot) * S0;
        out[0] = cost + 0.05f * bdr + 0.1f * tex;
    }
}

extern "C" void kernel_launch(void* const* d_in, const int* in_sizes, int n_in,
                              void* d_out, int out_size, void* d_ws, size_t ws_size,
                              hipStream_t stream) {
    const float* inp = (const float*)d_in[0];   // inputs  (16,1,512,512) f32
    const float* tgt = (const float*)d_in[1];   // targets (16,1,512,512) f32
    float* out = (float*)d_out;                 // scalar f32
    float* partial = (float*)d_ws;              // 4096 blocks * 8 floats

    dim3 grid(IMG / TILE, IMG / TILE, 16);      // 16 x 16 x 16 = 4096 blocks
    rcf_tile_kernel<<<grid, dim3(256), 0, stream>>>(inp, tgt, partial);
    rcf_finalize<<<dim3(1), dim3(256), 0, stream>>>(partial,
                                                    (IMG / TILE) * (IMG / TILE) * 16,
                                                    out);
}